// TriplanarGrid_5669356832254
// MI455X (gfx1250) — compile-verified
//
#include <hip/hip_runtime.h>

#define FDIM 16

#define AS1 __attribute__((address_space(1)))
#define AS3 __attribute__((address_space(3)))

#if defined(__AMDGCN__) && __has_builtin(__builtin_amdgcn_global_load_async_to_lds_b32)
#define HAVE_ASYNC_LDS 1
#else
#define HAVE_ASYNC_LDS 0
#endif

// ---------------------------------------------------------------------------
// Geometry helpers (match jax reference exactly)
// ---------------------------------------------------------------------------
__device__ __forceinline__ float reflectf(float v, float span) {
    float a = fabsf(v);
    float m = fmodf(a, 2.0f * span);
    return (m > span) ? (2.0f * span - m) : m;
}

__device__ __forceinline__ void bilin_setup(float gx, float gy, int S,
                                            int& i00, int& i01, int& i10, int& i11,
                                            float& w00, float& w01, float& w10, float& w11) {
    const float span = (float)(S - 1);
    float x = reflectf((gx + 1.0f) * 0.5f * span, span);
    float y = reflectf((gy + 1.0f) * 0.5f * span, span);
    float x0f = floorf(x);
    float y0f = floorf(y);
    float wx = x - x0f;
    float wy = y - y0f;
    int x0 = (int)x0f; if (x0 < 0) x0 = 0; if (x0 > S - 1) x0 = S - 1;
    int y0 = (int)y0f; if (y0 < 0) y0 = 0; if (y0 > S - 1) y0 = S - 1;
    int x1 = x0 + 1;   if (x1 > S - 1) x1 = S - 1;
    int y1 = y0 + 1;   if (y1 > S - 1) y1 = S - 1;
    i00 = y0 * S + x0;
    i01 = y0 * S + x1;
    i10 = y1 * S + x0;
    i11 = y1 * S + x1;
    w00 = (1.0f - wx) * (1.0f - wy);
    w01 = wx * (1.0f - wy);
    w10 = (1.0f - wx) * wy;
    w11 = wx * wy;
}

__device__ __forceinline__ void fma4(float4& a, float w, const float4 v) {
    a.x = fmaf(w, v.x, a.x);
    a.y = fmaf(w, v.y, a.y);
    a.z = fmaf(w, v.z, a.z);
    a.w = fmaf(w, v.w, a.w);
}

__device__ __forceinline__ void lod_params(int l, int& S, int& OFF) {
    switch (l) {
        case 0:  S = 129;  OFF = 0;        break;
        case 1:  S = 257;  OFF = 798768;   break;
        case 2:  S = 513;  OFF = 3969120;  break;
        default: S = 1025; OFF = 16601232; break;
    }
}

// ---------------------------------------------------------------------------
// Pre-pass: re-tile one plane [C,H,W] -> [H,W,C] so a corner's 16 channels
// become one contiguous 64B block (4x global_load_b128 in the sampler).
// Reads coalesced per channel; writes 64B-contiguous per thread (coalesced).
// ---------------------------------------------------------------------------
__global__ void triplane_transpose_kernel(const float* __restrict__ in,
                                          float* __restrict__ outp, int SS) {
    int i = blockIdx.x * blockDim.x + threadIdx.x;
    if (i >= SS) return;
    float v[FDIM];
#pragma unroll
    for (int c = 0; c < FDIM; ++c) v[c] = in[(size_t)c * SS + i];
    float4* o = reinterpret_cast<float4*>(outp + (size_t)i * FDIM);
    o[0] = make_float4(v[0],  v[1],  v[2],  v[3]);
    o[1] = make_float4(v[4],  v[5],  v[6],  v[7]);
    o[2] = make_float4(v[8],  v[9],  v[10], v[11]);
    o[3] = make_float4(v[12], v[13], v[14], v[15]);
}

// ---------------------------------------------------------------------------
// Main sampler over re-tiled [H,W,16] planes.
// One thread = (point, plane). Coordinates for the block's point window are
// staged into LDS with async global->LDS copies (ASYNCcnt path) so the 3
// plane-threads of a point don't re-issue redundant global loads. Then:
// 4 LODs x 4 corners x 4 float4 gathers = 64 independent b128 loads per
// thread; 16-float accumulator; one coalesced 64B store per thread.
// ---------------------------------------------------------------------------
__global__ void triplane_sample_hwc_kernel(const float* __restrict__ coords,
                                           const float* __restrict__ T,
                                           const int* __restrict__ lodp,
                                           float* __restrict__ out, int P) {
    __shared__ float scoords[264];

    const int tid0 = blockIdx.x * blockDim.x;
    const int tid  = tid0 + threadIdx.x;
    const int p0   = tid0 / 3;                  // first point covered by block
    int pLast = (tid0 + (int)blockDim.x - 1) / 3;
    if (pLast > P - 1) pLast = P - 1;
    const int span = (pLast - p0 + 1) * 3;      // <= 258 floats

#if HAVE_ASYNC_LDS
    for (int i = threadIdx.x; i < span; i += blockDim.x) {
        // builtin prototype (per hipcc diagnostic): (AS1 int*, AS3 int*, imm, imm)
        AS1 int* g = (AS1 int*)(uintptr_t)(coords + (size_t)p0 * 3 + i);
        // generic LDS address carries the LDS byte offset in its low 32 bits
        AS3 int* l = (AS3 int*)(unsigned)(uintptr_t)(&scoords[i]);
        __builtin_amdgcn_global_load_async_to_lds_b32(g, l, 0, 0);
    }
#if __has_builtin(__builtin_amdgcn_s_wait_asynccnt)
    __builtin_amdgcn_s_wait_asynccnt(0);
#else
    asm volatile("s_wait_asynccnt 0x0" ::: "memory");
#endif
#else
    for (int i = threadIdx.x; i < span; i += blockDim.x)
        scoords[i] = coords[(size_t)p0 * 3 + i];
#endif
    __syncthreads();

    if (tid >= P * 3) return;
    int p  = tid / 3;
    int pl = tid - p * 3;
    const int rp = p - p0;

    float cx = scoords[rp * 3 + 0];
    float cy = scoords[rp * 3 + 1];
    float cz = scoords[rp * 3 + 2];
    float gx = (pl == 0) ? cy : cx;
    float gy = (pl == 2) ? cy : cz;

    int nl = lodp[0];
    nl = (nl < 0) ? 0 : ((nl > 3) ? 3 : nl);
    nl += 1;

    float4 acc[4];
#pragma unroll
    for (int q = 0; q < 4; ++q) acc[q] = make_float4(0.f, 0.f, 0.f, 0.f);

    for (int l = 0; l < nl; ++l) {
        int S, OFF;
        lod_params(l, S, OFF);
        const float* base = T + (size_t)OFF + (size_t)pl * FDIM * S * S;

        int i00, i01, i10, i11;
        float w00, w01, w10, w11;
        bilin_setup(gx, gy, S, i00, i01, i10, i11, w00, w01, w10, w11);

        const float4* p00 = reinterpret_cast<const float4*>(base) + (size_t)i00 * 4;
        const float4* p01 = reinterpret_cast<const float4*>(base) + (size_t)i01 * 4;
        const float4* p10 = reinterpret_cast<const float4*>(base) + (size_t)i10 * 4;
        const float4* p11 = reinterpret_cast<const float4*>(base) + (size_t)i11 * 4;

#pragma unroll
        for (int q = 0; q < 4; ++q) {
            float4 a = p00[q];
            float4 b = p01[q];
            float4 c = p10[q];
            float4 d = p11[q];
            fma4(acc[q], w00, a);
            fma4(acc[q], w01, b);
            fma4(acc[q], w10, c);
            fma4(acc[q], w11, d);
        }
    }

    float4* o = reinterpret_cast<float4*>(out + (size_t)p * 48 + (size_t)pl * FDIM);
#pragma unroll
    for (int q = 0; q < 4; ++q) o[q] = acc[q];
}

// ---------------------------------------------------------------------------
// Fallback: direct gather from original [C,H,W] layout (used only when the
// workspace is too small for the re-tiled copy). Same math, strided channels.
// ---------------------------------------------------------------------------
struct Planes { const float* p[12]; };

__global__ void triplane_sample_chw_kernel(const float* __restrict__ coords,
                                           Planes ps,
                                           const int* __restrict__ lodp,
                                           float* __restrict__ out, int P) {
    int tid = blockIdx.x * blockDim.x + threadIdx.x;
    if (tid >= P * 3) return;
    int p  = tid / 3;
    int pl = tid - p * 3;

    float cx = coords[3 * p + 0];
    float cy = coords[3 * p + 1];
    float cz = coords[3 * p + 2];
    float gx = (pl == 0) ? cy : cx;
    float gy = (pl == 2) ? cy : cz;

    int nl = lodp[0];
    nl = (nl < 0) ? 0 : ((nl > 3) ? 3 : nl);
    nl += 1;

    float acc[FDIM];
#pragma unroll
    for (int c = 0; c < FDIM; ++c) acc[c] = 0.f;

    for (int l = 0; l < nl; ++l) {
        int S, OFF;
        lod_params(l, S, OFF);
        (void)OFF;
        const float* fm = ps.p[l * 3 + pl];
        int SS = S * S;

        int i00, i01, i10, i11;
        float w00, w01, w10, w11;
        bilin_setup(gx, gy, S, i00, i01, i10, i11, w00, w01, w10, w11);

#pragma unroll
        for (int c = 0; c < FDIM; ++c) {
            const float* b = fm + (size_t)c * SS;
            float r = fmaf(w00, b[i00], 0.f);
            r = fmaf(w01, b[i01], r);
            r = fmaf(w10, b[i10], r);
            r = fmaf(w11, b[i11], r);
            acc[c] += r;
        }
    }

    float* o = out + (size_t)p * 48 + (size_t)pl * FDIM;
#pragma unroll
    for (int c = 0; c < FDIM; ++c) o[c] = acc[c];
}

// ---------------------------------------------------------------------------
// Launch
// ---------------------------------------------------------------------------
extern "C" void kernel_launch(void* const* d_in, const int* in_sizes, int n_in,
                              void* d_out, int out_size, void* d_ws, size_t ws_size,
                              hipStream_t stream) {
    const float* coords = (const float*)d_in[0];
    const int*   lodp   = (const int*)d_in[13];
    float*       out    = (float*)d_out;
    const int P = in_sizes[0] / 3;

    static const int       SZh[4]  = {129, 257, 513, 1025};
    static const long long OFFh[4] = {0, 798768, 3969120, 16601232};
    const size_t need = 67031232ull * sizeof(float);  // full HWC re-tile of all 12 planes

    const long long n = (long long)P * 3;
    const int threads = 256;

    if (ws_size >= need) {
        float* T = (float*)d_ws;
        for (int l = 0; l < 4; ++l) {
            const int S  = SZh[l];
            const int SS = S * S;
            for (int pp = 0; pp < 3; ++pp) {
                const float* src = (const float*)d_in[1 + l * 3 + pp];
                float* dst = T + OFFh[l] + (long long)pp * FDIM * SS;
                triplane_transpose_kernel<<<(SS + threads - 1) / threads, threads, 0, stream>>>(
                    src, dst, SS);
            }
        }
        triplane_sample_hwc_kernel<<<(int)((n + threads - 1) / threads), threads, 0, stream>>>(
            coords, T, lodp, out, P);
    } else {
        Planes ps;
        for (int i = 0; i < 12; ++i) ps.p[i] = (const float*)d_in[1 + i];
        triplane_sample_chw_kernel<<<(int)((n + threads - 1) / threads), threads, 0, stream>>>(
            coords, ps, lodp, out, P);
    }
}